// FlowRegression_85907935855286
// MI455X (gfx1250) — compile-verified
//
#include <hip/hip_runtime.h>
#include <cstdint>

// Problem constants (from the reference)
constexpr int B  = 8;
constexpr int H  = 128;
constexpr int W  = 160;
constexpr int U  = 13;     // 2*MAXU+1
constexpr int V  = 13;     // 2*MAXV+1
constexpr int UV = U * V;  // 169
constexpr int MAXU  = 6;
constexpr int MAXV  = 6;
constexpr int TRUNC = 4;

constexpr int TILE_W = 32;              // one wave32 per block, one pixel per lane
constexpr int HW     = H * W;           // uv-plane stride in elements (20480)

typedef __attribute__((ext_vector_type(4))) unsigned int v4u;
typedef __attribute__((ext_vector_type(8))) int          v8i;
typedef __attribute__((ext_vector_type(4))) int          v4i;

__global__ __launch_bounds__(TILE_W)
void FlowRegression_tdm_kernel(const float* __restrict__ x, float* __restrict__ out) {
    // LDS tile: [uv][lane] -> lane reads its own column, bank-conflict free
    __shared__ float smem[UV * TILE_W];

    const int bid = blockIdx.x;
    const int wc  = bid % (W / TILE_W);
    const int h   = (bid / (W / TILE_W)) % H;
    const int b   = bid / ((W / TILE_W) * H);
    const int w0  = wc * TILE_W;
    const int t   = threadIdx.x;

    // LDS byte offset of smem: low 32 bits of the flat address of a __shared__
    // object are the LDS offset (shared aperture base has zero low 32 bits).
    // This ptrtoint flows into the TDM descriptor -> smem's address escapes, so
    // the optimizer must assume the DMA writes it (keeps the LDS loads alive).
    const uint32_t lds_off = (uint32_t)(uintptr_t)(void*)smem;

    // -------- Tensor DMA: 2D tile (32 contiguous pixels) x (169 uv planes) --------
    const uint64_t gaddr =
        (uint64_t)x + 4ull * ((uint64_t)b * (uint64_t)(UV * HW) + (uint64_t)h * W + (uint64_t)w0);

    // D# group 0 (ISA 8.3): count=1 | lds_addr | global_addr[56:0] | type=2
    v4u g0;
    g0.x = 1u;                                         // count=1, is_restore=0, gather off
    g0.y = lds_off;                                    // LDS byte address of tile start
    g0.z = (uint32_t)(gaddr & 0xFFFFFFFFull);          // global_addr[31:0]
    g0.w = (uint32_t)((gaddr >> 32) & 0x1FFFFFFull)    // global_addr[56:32]
         | (2u << 30);                                 // type = 2 ("image")

    // D# group 1 (ISA 8.4)
    v8i g1;
    g1[0] = (int)(2u << 16);                           // workgroup_mask=0, data_size=2 (4 bytes)
    g1[1] = (int)((uint32_t)TILE_W << 16);             // atomic_barrier=0 | tensor_dim0[15:0]=32
    g1[2] = (int)((uint32_t)UV << 16);                 // tensor_dim0[31:16]=0 | tensor_dim1[15:0]=169
    g1[3] = (int)((uint32_t)TILE_W << 16);             // tensor_dim1[31:16]=0 | tile_dim0=32
    g1[4] = (int)UV;                                   // tile_dim1=169 | tile_dim2=0
    g1[5] = (int)HW;                                   // tensor_dim0_stride[31:0] = 20480 elements
    g1[6] = 0;                                         // stride0[47:32]=0 | stride1[15:0]=0
    g1[7] = 0;                                         // stride1[47:16]=0
    v4i g2 = {0, 0, 0, 0};                             // 2D tensor: groups 2/3 unused (tile_dim3/4 = 0)
    v4i g3 = {0, 0, 0, 0};
    v8i g4 = {0, 0, 0, 0, 0, 0, 0, 0};                 // extra group (clang-23 6-arg form), unused

    __builtin_amdgcn_tensor_load_to_lds(g0, g1, g2, g3, g4, /*cpol*/0);
    __builtin_amdgcn_s_wait_tensorcnt(0);
    // Ordering fence only (the escape already happened via the descriptor):
    // LDS reads below must not hoist above the tensorcnt wait.
    asm volatile("" :: "s"(lds_off) : "memory");

    // -------- Pass 1: argmax over 169 channels (first-occurrence, like jnp.argmax) --------
    float m  = -3.402823466e38f;
    int   am = 0;
#pragma unroll
    for (int j = 0; j < UV; ++j) {
        float val = smem[j * TILE_W + t];           // ds_load_b32, bank-conflict free
        if (val > m) { m = val; am = j; }
    }

    // -------- Pass 2: softmax over the clamped +/-TRUNC window around the argmax --------
    const int u0  = am / V;
    const int v0  = am - u0 * V;
    const int ulo = (u0 - TRUNC > 0) ? u0 - TRUNC : 0;
    const int uhi = (u0 + TRUNC < U - 1) ? u0 + TRUNC : U - 1;
    const int vlo = (v0 - TRUNC > 0) ? v0 - TRUNC : 0;
    const int vhi = (v0 + TRUNC < V - 1) ? v0 + TRUNC : V - 1;

    float s = 0.0f, su = 0.0f, sv = 0.0f;
    for (int u = ulo; u <= uhi; ++u) {
        const float du   = (float)(u - MAXU);
        const int   base = u * V * TILE_W + t;
        for (int v = vlo; v <= vhi; ++v) {
            const float e = __expf(smem[base + v * TILE_W] - m);
            s  += e;
            su += e * du;
            sv += e * (float)(v - MAXV);
        }
    }
    const float inv = 1.0f / s;

    const int w = w0 + t;
    out[(((size_t)b * 2 + 0) * H + h) * W + w] = su * inv;
    out[(((size_t)b * 2 + 1) * H + h) * W + w] = sv * inv;
}

extern "C" void kernel_launch(void* const* d_in, const int* in_sizes, int n_in,
                              void* d_out, int out_size, void* d_ws, size_t ws_size,
                              hipStream_t stream) {
    const float* x   = (const float*)d_in[0];
    float*       out = (float*)d_out;
    (void)in_sizes; (void)n_in; (void)out_size; (void)d_ws; (void)ws_size;

    const int grid = B * H * (W / TILE_W);   // 5120 blocks, 1 wave32 each
    FlowRegression_tdm_kernel<<<grid, TILE_W, 0, stream>>>(x, out);
}